// DenoiseTVChambolleTransform_54631984005365
// MI455X (gfx1250) — compile-verified
//
#include <hip/hip_runtime.h>

// ---------------------------------------------------------------------------
// Persistent fused TV-Chambolle solver for MI455X (gfx1250).
//  - 3 channels x 16 blocks; each block keeps its 32x512 slab of p0/p1/img/d
//    resident in LDS (266KB of the WGP's 320KB) for all 200 iterations.
//  - TDM tensor_load_to_lds stages the (constant) image tile once.
//  - Cross-block sync: agent-scope atomic grid barrier (2 per iteration).
//  - Energy reduction per wave via v_wmma_f32_16x16x4_f32 (ones-selector B).
//  - Fast hardware rcp/sqrt (+1 Newton step) in the dual update.
// ---------------------------------------------------------------------------

typedef float        v2f __attribute__((ext_vector_type(2)));
typedef float        v8f __attribute__((ext_vector_type(8)));
typedef unsigned int v4u __attribute__((ext_vector_type(4)));
typedef int          v4i __attribute__((ext_vector_type(4)));
typedef int          v8i __attribute__((ext_vector_type(8)));

#define HH     512
#define WW     512
#define NCH    3
#define NBLK   16          // slabs per channel
#define ROWS   32          // rows per slab
#define NTHR   1024
#define NITER  200
#define WEIGHTF 0.1f
#define EPSF    2e-4f
#define TAUF    0.25f      // 1/(2*ndim)
#define TOWF    2.5f       // tau/weight

// workspace layout (bytes):
//   [0,64)        : int bar[16]        (per-channel barrier counters)
//   [64,128)      : int done[16]       (per-channel convergence flags)
//   [128,512)     : float part[3][16][2]
//   [1024, ...)   : float halos[2][3][3][16][512]  (slot, type, ch, blk)
#define HALO_FLOATS (2*3*3*NBLK*WW)
#define WS_ZERO_BYTES (1024 + (size_t)HALO_FLOATS*4)

__device__ __forceinline__ float* haloPtr(float* base, int slot, int type, int c, int b) {
    return base + ((((slot * 3) + type) * 3 + c) * NBLK + b) * WW;
}

// fast reciprocal: v_rcp_f32 + one Newton-Raphson step (~1 ulp; denom >= 1 here)
__device__ __forceinline__ float fastrcp(float x) {
    float r = __builtin_amdgcn_rcpf(x);
    return r * __builtin_fmaf(-x, r, 2.f);
}

__device__ __forceinline__ void gridbar(int* cnt, int target) {
    __builtin_amdgcn_fence(__ATOMIC_RELEASE, "agent");   // flush this block's stores
    __syncthreads();
    if (threadIdx.x == 0) {
        __hip_atomic_fetch_add(cnt, 1, __ATOMIC_RELAXED, __HIP_MEMORY_SCOPE_AGENT);
        while (__hip_atomic_load(cnt, __ATOMIC_RELAXED, __HIP_MEMORY_SCOPE_AGENT) < target)
            __builtin_amdgcn_s_sleep(2);
    }
    __syncthreads();
    __builtin_amdgcn_fence(__ATOMIC_ACQUIRE, "agent");   // invalidate stale WGP$ lines
}

__global__ __launch_bounds__(NTHR)
void tv_chambolle_persistent(const float* __restrict__ img,
                             float* __restrict__ out,
                             char* __restrict__ ws) {
    const int b   = blockIdx.x;    // slab
    const int c   = blockIdx.y;    // channel
    const int tid = threadIdx.x;
    const int r0  = b * ROWS;

    int*   bar   = (int*)ws;
    int*   done  = (int*)(ws + 64);
    float* part  = (float*)(ws + 128);
    float* halos = (float*)(ws + 1024);

    extern __shared__ float smem[];
    float* p0s  = smem;             // rows -1..32  -> 34*WW  (row i at (i+1)*WW)
    float* p1s  = smem + 34 * WW;   // rows  0..32  -> 33*WW
    float* dls  = smem + 67 * WW;   // rows  0..32  -> 33*WW
    float* imgs = smem + 100 * WW;  // rows  0..32  -> 33*WW
    float* reds = smem + 133 * WW;  // 66 floats (32 waves * 2 + stop flag)

    const float* gimg = img + (size_t)c * HH * WW + (size_t)r0 * WW;
    float*       gout = out + (size_t)c * HH * WW;

    // zero p0 (incl. halo rows) and p1 — contiguous 67*WW floats
    for (int idx = tid; idx < 67 * WW; idx += NTHR) p0s[idx] = 0.f;

    // --- TDM: DMA the image tile (33 rows x 512 cols, stride 512) into LDS.
    // tensor_dim1 = HH - r0 so the last slab's 33rd row reads OOB -> zeros.
    if (__builtin_amdgcn_readfirstlane((int)tid) == 0) {  // one wave issues
        unsigned long long ga = (unsigned long long)(const void*)gimg;
        v4u g0;
        g0[0] = 1u;                                        // count=1, user desc
        g0[1] = (unsigned)(100 * WW * 4);                  // lds_addr = imgs offset
        g0[2] = (unsigned)(ga & 0xffffffffull);            // global_addr lo
        g0[3] = (unsigned)((ga >> 32) & 0x01ffffffull) | 0x80000000u; // hi | type=2
        v8i g1;
        g1[0] = (int)(2u << 16);                           // data_size = 4B
        g1[1] = (int)((unsigned)WW << 16);                 // tensor_dim0 = 512
        g1[2] = (int)(((unsigned)(HH - r0)) << 16);        // tensor_dim1 = rows left
        g1[3] = (int)((unsigned)WW << 16);                 // tile_dim0 = 512
        g1[4] = 33;                                        // tile_dim1 = 33, tile_dim2=0
        g1[5] = WW;                                        // tensor_dim0_stride = 512
        g1[6] = 0;  g1[7] = 0;                             // tensor_dim1_stride = 0 (2D)
        v4i g2; g2[0] = 1; g2[1] = 1; g2[2] = 0; g2[3] = 0; // tensor_dim2/3 = 1
        v4i g3; g3[0] = 0; g3[1] = (1 << 16); g3[2] = 0; g3[3] = 0; // tensor_dim4 = 1
        v8i g4; g4[0] = 0; g4[1] = 0; g4[2] = 0; g4[3] = 0;         // extra group
        g4[4] = 0; g4[5] = 0; g4[6] = 0; g4[7] = 0;                 // (clang-23 form)
        __builtin_amdgcn_tensor_load_to_lds(g0, g1, g2, g3, g4, 0);
        __builtin_amdgcn_s_wait_tensorcnt(0);
    }
    __syncthreads();

    int   phase  = 0;
    float E_prev = 0.f, E_init = 0.f;

    for (int iter = 0; iter < NITER; ++iter) {
        const int rs  = 1 - (iter & 1);   // read  halo slot (prev iteration)
        const int wsl = iter & 1;         // write halo slot

        // (1) pull neighbor halo rows into LDS (slot rs zero-filled by memset for iter 0)
        if (tid < WW) {
            const int j = tid;
            if (b > 0)        p0s[j]            = haloPtr(halos, rs, 1, c, b - 1)[j]; // p0 row -1
            if (b < NBLK - 1) {
                p0s[33 * WW + j] = haloPtr(halos, rs, 0, c, b + 1)[j];                // p0 row 32
                p1s[32 * WW + j] = haloPtr(halos, rs, 2, c, b + 1)[j];                // p1 row 32
            }
        }
        __syncthreads();

        // (2) d = div(p) for local rows 0..32
        for (int k = 0; k < 17; ++k) {
            const int idx = tid + k * NTHR;
            if (idx < 33 * WW) {
                const int i = idx >> 9, j = idx & (WW - 1);
                float v = -p0s[(i + 1) * WW + j] - p1s[i * WW + j] + p0s[i * WW + j];
                if (j > 0) v += p1s[i * WW + j - 1];
                dls[idx] = v;
            }
        }
        __syncthreads();

        // (3) gradient / norm / energy / in-place dual update (own pixels only)
        float e_d2 = 0.f, e_nrm = 0.f;
        for (int k = 0; k < 16; ++k) {
            const int idx = tid + k * NTHR;
            const int i = idx >> 9, j = idx & (WW - 1);
            const int gi = r0 + i;
            const float dc = dls[idx];
            const float ic = imgs[idx];
            float g0v = 0.f, g1v = 0.f;
            if (gi < HH - 1) g0v = (dls[idx + WW] + imgs[idx + WW]) - (dc + ic);
            if (j  < WW - 1) g1v = (dls[idx + 1]  + imgs[idx + 1])  - (dc + ic);
            const float nrm = __builtin_amdgcn_sqrtf(__builtin_fmaf(g0v, g0v, g1v * g1v));
            e_d2 = __builtin_fmaf(dc, dc, e_d2);
            e_nrm += nrm;
            const float inv = fastrcp(__builtin_fmaf(TOWF, nrm, 1.f));
            p0s[(i + 1) * WW + j] = (p0s[(i + 1) * WW + j] - TAUF * g0v) * inv;
            p1s[i * WW + j]       = (p1s[i * WW + j]       - TAUF * g1v) * inv;
        }

        // (4) wave reduction with one v_wmma_f32_16x16x4_f32:
        //     A vgpr0 = d2[lane], vgpr1 = norm[lane]  (A[m][{0,1}]=lane m, A[m][{2,3}]=lane m+16)
        //     B = selector: col 0 sums K=0,1 (d2+norm, lanes 0-15), col 8 sums K=2,3 (d2+norm, lanes 16-31)
        {
            const int lane = tid & 31, wid = tid >> 5;
            v2f a;  a[0] = e_d2; a[1] = e_nrm;
            const float bs = (lane == 0 || lane == 24) ? 1.f : 0.f;
            v2f bb; bb[0] = bs; bb[1] = bs;
            v8f cz = {0.f, 0.f, 0.f, 0.f, 0.f, 0.f, 0.f, 0.f};
            v8f dv = __builtin_amdgcn_wmma_f32_16x16x4_f32(false, a, false, bb,
                                                           (short)0, cz, false, false);
            const float s = dv[0] + dv[1] + dv[2] + dv[3] + dv[4] + dv[5] + dv[6] + dv[7];
            const float d2t = __shfl(s, 0, 32) + __shfl(s, 16, 32);
            const float nt  = __shfl(s, 8, 32) + __shfl(s, 24, 32);
            if (lane == 0) { reds[2 * wid] = d2t; reds[2 * wid + 1] = nt; }
        }
        __syncthreads();

        // (5) publish updated boundary rows + block energy partial
        if (tid < WW) {
            const int j = tid;
            haloPtr(halos, wsl, 0, c, b)[j] = p0s[WW + j];        // p0 row 0
            haloPtr(halos, wsl, 1, c, b)[j] = p0s[32 * WW + j];   // p0 row 31
            haloPtr(halos, wsl, 2, c, b)[j] = p1s[j];             // p1 row 0
        }
        if (tid == 0) {
            float td2 = 0.f, tn = 0.f;
            for (int w = 0; w < NTHR / 32; ++w) { td2 += reds[2 * w]; tn += reds[2 * w + 1]; }
            part[(c * NBLK + b) * 2 + 0] = td2;
            part[(c * NBLK + b) * 2 + 1] = tn;
        }

        ++phase;
        gridbar(&bar[c], phase * NBLK);

        // (6) block 0 finalizes E and the early-stop decision (fixed-order sum)
        if (b == 0 && tid == 0) {
            float sd2 = 0.f, sn = 0.f;
            for (int q = 0; q < NBLK; ++q) {
                sd2 += part[(c * NBLK + q) * 2 + 0];
                sn  += part[(c * NBLK + q) * 2 + 1];
            }
            const float E = (sd2 + WEIGHTF * sn) * (1.f / (float)(HH * WW));
            if (iter == 0) E_init = E;
            else if (fabsf(E_prev - E) < EPSF * E_init)
                __hip_atomic_store(&done[c], 1, __ATOMIC_RELAXED, __HIP_MEMORY_SCOPE_AGENT);
            E_prev = E;
        }

        ++phase;
        gridbar(&bar[c], phase * NBLK);

        // (7) stop? out = img + div(p_before_update) == out_new of the conv step
        if (tid == 0)
            ((int*)reds)[64] = __hip_atomic_load(&done[c], __ATOMIC_RELAXED,
                                                 __HIP_MEMORY_SCOPE_AGENT);
        __syncthreads();
        const int stop = ((int*)reds)[64];
        if (stop || iter == NITER - 1) {
            for (int k = 0; k < 16; ++k) {
                const int idx = tid + k * NTHR;
                gout[(size_t)r0 * WW + idx] = dls[idx] + imgs[idx];
            }
            break;
        }
    }
}

extern "C" void kernel_launch(void* const* d_in, const int* in_sizes, int n_in,
                              void* d_out, int out_size, void* d_ws, size_t ws_size,
                              hipStream_t stream) {
    (void)in_sizes; (void)n_in; (void)out_size; (void)ws_size;
    const float* img = (const float*)d_in[0];
    float* out = (float*)d_out;

    // zero barrier counters / done flags / halo slot buffers (graph-capturable)
    (void)hipMemsetAsync(d_ws, 0, WS_ZERO_BYTES, stream);

    dim3 grid(NBLK, NCH);
    size_t smem = (size_t)(133 * WW + 66) * sizeof(float);   // ~266 KB of 320 KB LDS
    tv_chambolle_persistent<<<grid, dim3(NTHR), smem, stream>>>(img, out, (char*)d_ws);
}